// DynamicParamNetwork_61194694033740
// MI455X (gfx1250) — compile-verified
//
#include <hip/hip_runtime.h>
#include <hip/hip_bf16.h>
#include <math.h>

// Problem dims (fixed by the reference)
#define B_DIM   256
#define IN_DIM  512
#define H_DIM   512
#define OUT_DIM 256
#define KF      10              // NUM_CENTERS * F = 5 * 2
#define N1      (H_DIM * KF)    // 5120  (w1 flattened [IN, H*KF])
#define N2      (OUT_DIM * KF)  // 2560  (w2 flattened [H, OUT*KF])

// GEMM tiling
#define KT      16              // K rows staged per LDS buffer
#define NW      160             // n-columns per block (= 16 output cols * KF)
#define KTP     20              // LDS column stride in floats (16 + pad 4):
                                //  (20*col)%64 -> 16 distinct bank groups of 4
#define STAGEF  (NW * KTP)      // floats per stage buffer (transposed [col][k])

typedef __attribute__((ext_vector_type(2))) float v2f;
typedef __attribute__((ext_vector_type(8))) float v8f;

// ---------------------------------------------------------------------------
// gfx1250 async global->LDS copies (ASYNCcnt-tracked).
// VDST = LDS byte address VGPR, VADDR = 64-bit global address VGPR pair.
// ---------------------------------------------------------------------------
__device__ __forceinline__ unsigned lds_off_u32(const void* p) {
    // Generic pointers to LDS carry the byte offset in the low 32 bits.
    return (unsigned)(uintptr_t)p;
}

__device__ __forceinline__ void async_copy_b32(unsigned lds_addr, const void* gaddr) {
    asm volatile("global_load_async_to_lds_b32 %0, %1, off"
                 :: "v"(lds_addr), "v"(gaddr)
                 : "memory");
}

__device__ __forceinline__ void wait_async_le20() {
    asm volatile("s_wait_asynccnt 0x14" ::: "memory");
}
__device__ __forceinline__ void wait_async_0() {
    asm volatile("s_wait_asynccnt 0x0" ::: "memory");
}

// ---------------------------------------------------------------------------
// Kernel A: multiquadric RBF basis.  basis[b, j*2+i] = sqrt((a[b,i]-c_j)^2+1)
// ---------------------------------------------------------------------------
__global__ __launch_bounds__(256) void basis_kernel(const float* __restrict__ a,
                                                    float* __restrict__ basis) {
    int b = blockIdx.x * blockDim.x + threadIdx.x;
    if (b >= B_DIM) return;
    float a0 = a[b * 2 + 0];
    float a1 = a[b * 2 + 1];
#pragma unroll
    for (int j = 0; j < 5; ++j) {
        float c = -1.0f + 0.5f * (float)j;
        float d0 = a0 - c, d1 = a1 - c;
        basis[b * KF + j * 2 + 0] = sqrtf(d0 * d0 + 1.0f);
        basis[b * KF + j * 2 + 1] = sqrtf(d1 * d1 + 1.0f);
    }
}

// ---------------------------------------------------------------------------
// Fused GEMM + basis-fold.
//
//   C[b, n] = sum_i A[b, i] * W[i, n]        (W: [Kred x NMAT] row-major)
//   out[b, c] = (relu?)( sum_k (C[b, c*10+k] (+ bias[c,k])) * basis[b,k] )
//
// Block = 128 threads = 4 waves.  M_block = 64 (wave w owns b-rows
// b0+16w..+15); all 4 waves share one 160-column W slab, staged TRANSPOSED
// ([col][k], col stride KTP) in LDS by async b32 copies, double-buffered over
// K in steps of KT=16.  Transposed layout makes every WMMA B-fragment a
// single aligned ds_load_b64 (K-pair contiguous) -- no register shuffles.
// WMMA: V_WMMA_F32_16X16X4_F32, 10 accumulators (16x16) per wave.
// ---------------------------------------------------------------------------
template <int NMAT, bool RELU, bool HAS_BIAS>
__device__ __forceinline__ void gemm_fold(const float* __restrict__ A,
                                          const float* __restrict__ W,
                                          const float* __restrict__ bias,   // [ncols, KF]
                                          const float* __restrict__ basis,  // [B_DIM, KF]
                                          float* __restrict__ out,          // [B_DIM, ncols]
                                          int ncols_total,
                                          int Kred) {
    __shared__ float wstage[2 * STAGEF];      // 2 x 12.8 KB transposed W stages
    __shared__ float cspill[4 * 16 * NW];     // 40 KB C spill for the fold

    const int tid   = threadIdx.x;
    const int lane  = tid & 31;
    const int wave  = tid >> 5;
    const int btile = blockIdx.x & 3;         // 4 tiles of 64 rows over B
    const int cblk  = blockIdx.x >> 2;        // blocks of 16 output columns
    const int b0    = btile * 64;
    const int bw    = b0 + wave * 16;         // this wave's 16 b-rows
    const int cbase = cblk * 16;              // output column range
    const int n0    = cbase * KF;             // 160-wide n range (shared by block)

    const int row   = lane & 15;              // M (or N) index inside a 16 tile
    const int khalf = (lane >> 4) * 2;        // 0 or 2 (A/B fragment K split)

    // --- async staging: 2560 floats, 20 per thread, written transposed ---
    const unsigned lds_base0 = lds_off_u32(&wstage[0]);
    auto issue_stage = [&](int kk, int bufsel) {
        const float* gbase = W + (size_t)kk * NMAT + n0;
        unsigned lbase = lds_base0 + (unsigned)(bufsel * STAGEF * 4);
#pragma unroll
        for (int j = 0; j < 20; ++j) {
            int e = tid + j * 128;            // 0..2559
            int k = e / NW;                   // stage row (K), 0..15
            int p = e - k * NW;               // n within slab, 0..159
            const float* gp = gbase + (size_t)k * NMAT + p;
            unsigned la = lbase + (unsigned)((p * KTP + k) * 4);
            async_copy_b32(la, gp);
        }
    };

    v8f acc[10] = {};

    const int nstages = Kred / KT;
    issue_stage(0, 0);

    for (int s = 0; s < nstages; ++s) {
        if (s + 1 < nstages) {
            issue_stage((s + 1) * KT, (s + 1) & 1);
            wait_async_le20();                // in-order => stage s complete
        } else {
            wait_async_0();
        }
        __syncthreads();                      // stage s visible to all waves

        const float* st   = wstage + (s & 1) * STAGEF;
        const float* arow = A + (size_t)(bw + row) * Kred + s * KT + khalf;

        // prefetch all 4 A-fragments of this stage (one clause of b64 loads)
        v2f af[4];
#pragma unroll
        for (int q = 0; q < 4; ++q) af[q] = *(const v2f*)(arow + 4 * q);

        const float* sp0 = st + row * KTP + khalf;  // lane base in transposed stage
#pragma unroll
        for (int q = 0; q < 4; ++q) {               // k-step = 4*q
            v2f afrag = af[q];
#pragma unroll
            for (int t = 0; t < 10; ++t) {
                // B-fragment: K-pair contiguous -> single ds_load_b64
                v2f bfrag = *(const v2f*)(sp0 + t * (16 * KTP) + q * 4);
                acc[t] = __builtin_amdgcn_wmma_f32_16x16x4_f32(
                    false, afrag, false, bfrag, (short)0, acc[t], false, false);
            }
        }
        __syncthreads();                      // readers done before buffer reuse
    }

    // --- spill slab: VGPR v, lane l -> M = v + 8*(l>>4), N = l&15 ---
    float* myl = cspill + wave * (16 * NW);
    const int mhi = 8 * (lane >> 4);
#pragma unroll
    for (int t = 0; t < 10; ++t) {
#pragma unroll
        for (int v = 0; v < 8; ++v) {
            myl[(v + mhi) * NW + t * 16 + row] = acc[t][v];
        }
    }
    __syncthreads();

    // --- fold groups of 10 with basis: 64 b x 16 cols = 1024 outputs ---
    for (int p = tid; p < 64 * 16; p += 128) {
        int b  = p >> 4;                      // 0..63
        int cg = p & 15;                      // 0..15 column within block
        const float* c   = cspill + (b >> 4) * (16 * NW) + (b & 15) * NW + cg * KF;
        const float* bas = basis + (b0 + b) * KF;
        float sacc = 0.0f;
        if (HAS_BIAS) {
            const float* bi = bias + (size_t)(cbase + cg) * KF;
#pragma unroll
            for (int k = 0; k < KF; ++k) sacc += (c[k] + bi[k]) * bas[k];
        } else {
#pragma unroll
            for (int k = 0; k < KF; ++k) sacc += c[k] * bas[k];
        }
        if (RELU) sacc = fmaxf(sacc, 0.0f);
        out[(size_t)(b0 + b) * ncols_total + (cbase + cg)] = sacc;
    }
}

__global__ __launch_bounds__(128) void gemm1_kernel(const float* __restrict__ x,
                                                    const float* __restrict__ w1,
                                                    const float* __restrict__ b1,
                                                    const float* __restrict__ basis,
                                                    float* __restrict__ hact) {
    gemm_fold<N1, true, true>(x, w1, b1, basis, hact, H_DIM, IN_DIM);
}

__global__ __launch_bounds__(128) void gemm2_kernel(const float* __restrict__ hact,
                                                    const float* __restrict__ w2,
                                                    const float* __restrict__ basis,
                                                    float* __restrict__ raw) {
    gemm_fold<N2, false, false>(hact, w2, nullptr, basis, raw, OUT_DIM, H_DIM);
}

// ---------------------------------------------------------------------------
// Kernel D: per-row standardization, std with ddof=1
// ---------------------------------------------------------------------------
__global__ __launch_bounds__(256) void norm_kernel(const float* __restrict__ raw,
                                                   float* __restrict__ out) {
    __shared__ float buf[OUT_DIM];
    const int b = blockIdx.x;
    const int t = threadIdx.x;
    float v = raw[b * OUT_DIM + t];
    buf[t] = v;
    __syncthreads();
    for (int s = OUT_DIM / 2; s > 0; s >>= 1) {
        if (t < s) buf[t] += buf[t + s];
        __syncthreads();
    }
    float mean = buf[0] * (1.0f / OUT_DIM);
    __syncthreads();
    float d = v - mean;
    buf[t] = d * d;
    __syncthreads();
    for (int s = OUT_DIM / 2; s > 0; s >>= 1) {
        if (t < s) buf[t] += buf[t + s];
        __syncthreads();
    }
    float stdv = sqrtf(buf[0] * (1.0f / (OUT_DIM - 1)));
    out[b * OUT_DIM + t] = d / (stdv + 1e-8f);
}

// ---------------------------------------------------------------------------
// Launch
// ---------------------------------------------------------------------------
extern "C" void kernel_launch(void* const* d_in, const int* in_sizes, int n_in,
                              void* d_out, int out_size, void* d_ws, size_t ws_size,
                              hipStream_t stream) {
    const float* x  = (const float*)d_in[0];   // [256, 1, 512] -> [256, 512]
    const float* a  = (const float*)d_in[1];   // [256, 2]
    const float* w1 = (const float*)d_in[2];   // [512, 512, 10] -> [512, 5120]
    const float* b1 = (const float*)d_in[3];   // [512, 10]
    const float* w2 = (const float*)d_in[4];   // [512, 256, 10] -> [512, 2560]
    float* out = (float*)d_out;                // [256, 256]

    float* ws    = (float*)d_ws;
    float* basis = ws;                         // 256*10  floats
    float* hact  = ws + 4096;                  // 256*512 floats
    float* raw   = ws + 4096 + 131072;         // 256*256 floats

    basis_kernel<<<1, 256, 0, stream>>>(a, basis);
    gemm1_kernel<<<4 * (H_DIM / 16), 128, 0, stream>>>(x, w1, b1, basis, hact);
    gemm2_kernel<<<4 * (OUT_DIM / 16), 128, 0, stream>>>(hact, w2, basis, raw);
    norm_kernel<<<B_DIM, OUT_DIM, 0, stream>>>(raw, out);
}